// AddingGaussianBlur_11562051961368
// MI455X (gfx1250) — compile-verified
//
#include <hip/hip_runtime.h>
#include <hip/hip_bf16.h>
#include <stdint.h>

// AddingGaussianBlur — per-batch 3x3 Gaussian, separable (weights depend only on column):
//   out[b,h,w,c] = sum_{di,dj} k[b,dj] * xpad[b,h+di,w+dj,c]
//   k[b,-1..1] = {w1, 1, w1} / (3*(1+2*w1)),  w1 = exp(-1/s^2),  s = stds[b]*3
//
// Memory-bound (402 MB traffic, ~0.9 GFLOP) -> use CDNA5 async global->LDS staging
// (global_load_async_to_lds_b128 / ASYNCcnt) + LDS-resident separable stencil.

#define FACTOR 3.0f

namespace {
constexpr int B     = 64;
constexpr int H     = 512;
constexpr int W     = 512;
constexpr int C     = 3;
constexpr int ROWF  = W * C;        // 1536 floats per image row
constexpr int ROWB  = ROWF * 4;     // 6144 bytes per image row
constexpr int CH16  = ROWF / 4;     // 384 float4 chunks per row
constexpr int TR    = 4;            // output rows per block
constexpr int TROWS = TR + 2;       // tile rows incl. halo
constexpr int NTHREADS = 256;       // 8 waves (wave32)
}

__global__ __launch_bounds__(NTHREADS)
void gauss3x3_kernel(const float* __restrict__ x,
                     const float* __restrict__ stds,
                     float* __restrict__ out)
{
    __shared__ float tile[TROWS * ROWF];   // 36,864 B : input rows (+halo)
    __shared__ float vsum[TR * ROWF];      // 24,576 B : vertical 3-row sums

    const int tid       = threadIdx.x;
    const int rowBlocks = H / TR;                       // 128
    const int b         = blockIdx.x / rowBlocks;
    const int r0        = (blockIdx.x % rowBlocks) * TR;

    // ---- per-batch weights (scalar-ish, negligible cost) ----
    const float s   = stds[b] * FACTOR;
    const float w1  = __expf(-1.0f / (s * s));
    const float inv = 1.0f / (3.0f * (1.0f + 2.0f * w1));
    const float wc  = inv;        // center column weight
    const float wn  = w1 * inv;   // +/-1 column weight

    // ---- async-stage TROWS rows of this image into LDS (per-lane B128 copies) ----
    // LDS byte address: low 32 bits of the generic pointer == wave-relative LDS offset.
    const unsigned ldsTile = (unsigned)(size_t)(void*)&tile[0];
    const unsigned rowBase = (unsigned)(b * H);          // row index of image b, row 0

    #pragma unroll
    for (int trr = 0; trr < TROWS; ++trr) {
        const int r = r0 - 1 + trr;
        if (r >= 0 && r < H) {
            const unsigned gRow = (rowBase + (unsigned)r) * (unsigned)ROWB; // < 2^31
            for (int c16 = tid; c16 < CH16; c16 += NTHREADS) {
                const unsigned goff  = gRow + (unsigned)(c16 * 16);
                const unsigned laddr = ldsTile + (unsigned)(trr * ROWB + c16 * 16);
                asm volatile("global_load_async_to_lds_b128 %0, %1, %2"
                             :
                             : "v"(laddr), "v"(goff), "s"(x)
                             : "memory");
            }
        }
    }
    asm volatile("s_wait_asynccnt 0" ::: "memory");
    __syncthreads();

    // ---- phase 1: vertical 3-row sum (zero-padded via predicated adds) ----
    const bool topValid = (r0 > 0);
    const bool botValid = (r0 + TR < H);

    #pragma unroll
    for (int tr = 0; tr < TR; ++tr) {
        const bool hasTop = (tr > 0)      || topValid;   // tile[tr]   is a real row
        const bool hasBot = (tr < TR - 1) || botValid;   // tile[tr+2] is a real row
        for (int c16 = tid; c16 < CH16; c16 += NTHREADS) {
            const int j0 = c16 * 4;
            float4 acc = *(const float4*)&tile[(tr + 1) * ROWF + j0];
            if (hasTop) {
                const float4 t = *(const float4*)&tile[tr * ROWF + j0];
                acc.x += t.x; acc.y += t.y; acc.z += t.z; acc.w += t.w;
            }
            if (hasBot) {
                const float4 t = *(const float4*)&tile[(tr + 2) * ROWF + j0];
                acc.x += t.x; acc.y += t.y; acc.z += t.z; acc.w += t.w;
            }
            *(float4*)&vsum[tr * ROWF + j0] = acc;
        }
    }
    __syncthreads();

    // ---- phase 2: horizontal [wn, wc, wn] over channels (+/-3 floats), coalesced stores ----
    #pragma unroll
    for (int tr = 0; tr < TR; ++tr) {
        const float* v = &vsum[tr * ROWF];
        const size_t gRow = (size_t)(b * H + r0 + tr) * (size_t)ROWF;
        for (int c16 = tid; c16 < CH16; c16 += NTHREADS) {
            const int j0 = c16 * 4;
            float4 o;
            #pragma unroll
            for (int e = 0; e < 4; ++e) {
                const int j = j0 + e;
                const float vl = (j >= 3)        ? v[j - 3] : 0.0f;
                const float vr = (j < ROWF - 3)  ? v[j + 3] : 0.0f;
                const float vcv = v[j];
                const float r = fmaf(wn, vl, fmaf(wn, vr, wc * vcv));
                if (e == 0) o.x = r; else if (e == 1) o.y = r;
                else if (e == 2) o.z = r; else o.w = r;
            }
            *(float4*)&out[gRow + (size_t)j0] = o;
        }
    }
}

extern "C" void kernel_launch(void* const* d_in, const int* in_sizes, int n_in,
                              void* d_out, int out_size, void* d_ws, size_t ws_size,
                              hipStream_t stream) {
    (void)in_sizes; (void)n_in; (void)d_ws; (void)ws_size; (void)out_size;
    const float* x    = (const float*)d_in[0];   // (64,512,512,3) f32
    const float* stds = (const float*)d_in[1];   // (64,) f32
    float* out        = (float*)d_out;

    const int grid = B * (H / TR);               // 8192 blocks
    gauss3x3_kernel<<<grid, NTHREADS, 0, stream>>>(x, stds, out);
}